// BlurContrastiveModelPair_2138893714020
// MI455X (gfx1250) — compile-verified
//
#include <hip/hip_runtime.h>
#include <hip/hip_bf16.h>

#define T_DIM 512
#define D_DIM 512
#define B_DIM 256

typedef __attribute__((ext_vector_type(2))) float v2f;
typedef __attribute__((ext_vector_type(8))) float v8f;

// K-pair-interleaved LDS layout: element (k, j) lives at
//   lds[(k>>1)*PAIR_STRIDE + j*2 + (k&1)]
// so a lane's {k, k+1} WMMA fragment is one aligned 8-byte ds_load_b64.
// PAIR_STRIDE has +32 floats of pad: consecutive pair-rows are shifted by
// 32 LDS banks, so lane halves 0-15 (pair p) / 16-31 (pair p+1) of a single
// b64 load use disjoint bank halves -> conflict-free.
#define PS_S 288   // 128 j * 2 + 32 pad
#define PS_Q 160   //  64 d * 2 + 32 pad

// avged_seq[b,j,d] = sum_t S[b,t,j] * seq[b,t,d]
// Block = 8 waves = 128(j) x 64(d) tile of batch b; each wave owns 32x32.
__global__ __launch_bounds__(256)
void blur_gemm_wmma(const float* __restrict__ S, const float* __restrict__ seq,
                    float* __restrict__ out /* avged_seq */) {
  __shared__ float ldsS[16 * PS_S];  // 32 k x 128 j (interleaved pairs)
  __shared__ float ldsQ[16 * PS_Q];  // 32 k x  64 d

  const int d0 = blockIdx.x * 64;
  const int j0 = blockIdx.y * 128;
  const int b  = blockIdx.z;

  const int tid  = threadIdx.x;
  const int lane = tid & 31;
  const int wave = tid >> 5;        // 0..7
  const int jw   = (wave & 3) * 32; // wave j base within tile
  const int dw   = (wave >> 2) * 32;// wave d base within tile

  const int half = lane >> 4;       // fp32 WMMA lane-half (K=0,1 vs K=2,3)
  const int m    = lane & 15;

  const float* Sb = S   + (size_t)b * T_DIM * T_DIM;
  const float* Qb = seq + (size_t)b * T_DIM * D_DIM;

  v8f c00 = {}, c01 = {}, c10 = {}, c11 = {};

  for (int t0 = 0; t0 < T_DIM; t0 += 32) {
    // ---- stage S[t0..t0+31][j0..j0+127] (4096 floats) interleaved ----
    #pragma unroll
    for (int i = 0; i < 4; ++i) {
      const int f = tid + 256 * i;        // float4 id 0..1023
      const int p = f >> 6;               // pair-row 0..15
      const int q = f & 63;               // j-pair 0..63 (j = 2q)
      const float2 g0 = *(const float2*)(Sb + (size_t)(t0 + 2 * p    ) * T_DIM + j0 + 2 * q);
      const float2 g1 = *(const float2*)(Sb + (size_t)(t0 + 2 * p + 1) * T_DIM + j0 + 2 * q);
      float4 v; v.x = g0.x; v.y = g1.x; v.z = g0.y; v.w = g1.y;
      *(float4*)(&ldsS[p * PS_S + 4 * q]) = v;
    }
    // ---- stage seq[t0..t0+31][d0..d0+63] (2048 floats) interleaved ----
    #pragma unroll
    for (int i = 0; i < 2; ++i) {
      const int f = tid + 256 * i;        // float4 id 0..511
      const int p = f >> 5;               // pair-row 0..15
      const int q = f & 31;               // d-pair 0..31
      const float2 g0 = *(const float2*)(Qb + (size_t)(t0 + 2 * p    ) * D_DIM + d0 + 2 * q);
      const float2 g1 = *(const float2*)(Qb + (size_t)(t0 + 2 * p + 1) * D_DIM + d0 + 2 * q);
      float4 v; v.x = g0.x; v.y = g1.x; v.z = g0.y; v.w = g1.y;
      *(float4*)(&ldsQ[p * PS_Q + 4 * q]) = v;
    }
    // Prefetch next K-chunk while computing this one.
    if (t0 + 32 < T_DIM) {
      __builtin_prefetch(Sb + (size_t)(t0 + 32 + (tid >> 3)) * T_DIM + j0 + (tid & 7) * 16, 0, 0);
      __builtin_prefetch(Qb + (size_t)(t0 + 32 + (tid >> 3)) * D_DIM + d0 + (tid & 3) * 16, 0, 0);
    }
    __syncthreads();

    // A[j, t] = S[t, j]; fragments per ISA 16x4 fp32 layout.
    #pragma unroll
    for (int kk = 0; kk < 32; kk += 4) {
      const int p = (kk >> 1) + half;     // pair-row for this lane half
      const v2f a0 = *(const v2f*)&ldsS[p * PS_S + (jw      + m) * 2];
      const v2f a1 = *(const v2f*)&ldsS[p * PS_S + (jw + 16 + m) * 2];
      const v2f b0 = *(const v2f*)&ldsQ[p * PS_Q + (dw      + m) * 2];
      const v2f b1 = *(const v2f*)&ldsQ[p * PS_Q + (dw + 16 + m) * 2];
      c00 = __builtin_amdgcn_wmma_f32_16x16x4_f32(false, a0, false, b0, (short)0, c00, false, false);
      c01 = __builtin_amdgcn_wmma_f32_16x16x4_f32(false, a0, false, b1, (short)0, c01, false, false);
      c10 = __builtin_amdgcn_wmma_f32_16x16x4_f32(false, a1, false, b0, (short)0, c10, false, false);
      c11 = __builtin_amdgcn_wmma_f32_16x16x4_f32(false, a1, false, b1, (short)0, c11, false, false);
    }
    __syncthreads();
  }

  // C/D layout: VGPR r -> row M = r + 8*half, col N = m; coalesced stores.
  #pragma unroll
  for (int r = 0; r < 8; ++r) {
    const int jrowA = j0 + jw + r + 8 * half;        // sub-tile row block 0
    const int jrowB = jrowA + 16;                    // sub-tile row block 1
    float* rowA = out + ((size_t)b * T_DIM + jrowA) * D_DIM + d0 + dw;
    float* rowB = out + ((size_t)b * T_DIM + jrowB) * D_DIM + d0 + dw;
    rowA[m]      = c00[r];
    rowA[16 + m] = c01[r];
    rowB[m]      = c10[r];
    rowB[16 + m] = c11[r];
  }
}

// Straight passthrough of SeqtoBlur (output 0): float4 streaming copy.
__global__ __launch_bounds__(256)
void copy_f4(const float4* __restrict__ in, float4* __restrict__ out, int n4) {
  const int i = blockIdx.x * blockDim.x + threadIdx.x;
  if (i < n4) out[i] = in[i];
}

// R[b,t] = (t+1)/max(len,1) if t < len else 0 ; same for avged_R with
// avged_len; also emit avged_len as float (tuple concatenated as f32).
__global__ __launch_bounds__(256)
void ratios_kernel(const int* __restrict__ len_seq, const int* __restrict__ avged_len,
                   float* __restrict__ R, float* __restrict__ aR,
                   float* __restrict__ alen_out) {
  const int idx = blockIdx.x * blockDim.x + threadIdx.x;  // B*T
  const int b = idx / T_DIM;
  const int t = idx % T_DIM;
  const float l  = (float)len_seq[b];
  const float al = (float)avged_len[b];
  const float tf = (float)t;
  R[idx]  = (tf < l)  ? (tf + 1.0f) / fmaxf(l, 1.0f)  : 0.0f;
  aR[idx] = (tf < al) ? (tf + 1.0f) / fmaxf(al, 1.0f) : 0.0f;
  if (t == 0) alen_out[b] = al;
}

extern "C" void kernel_launch(void* const* d_in, const int* in_sizes, int n_in,
                              void* d_out, int out_size, void* d_ws, size_t ws_size,
                              hipStream_t stream) {
  (void)in_sizes; (void)n_in; (void)out_size; (void)d_ws; (void)ws_size;

  const float* seq       = (const float*)d_in[0];  // [B,T,D]
  const float* SeqtoBlur = (const float*)d_in[1];  // [B,T,T]
  const int*   len_seq   = (const int*)d_in[2];    // [B]
  const int*   avged_len = (const int*)d_in[3];    // [B]

  // Output tuple, concatenated flat (float32):
  //   [0] SeqtoBlur  B*T*T
  //   [1] avged_seq  B*T*D
  //   [2] R          B*T
  //   [3] avged_R    B*T
  //   [4] avged_len  B
  const size_t szS = (size_t)B_DIM * T_DIM * T_DIM;
  const size_t szA = (size_t)B_DIM * T_DIM * D_DIM;
  const size_t szR = (size_t)B_DIM * T_DIM;

  float* outS  = (float*)d_out;
  float* outA  = outS + szS;
  float* outR  = outA + szA;
  float* outAR = outR + szR;
  float* outAL = outAR + szR;

  // Passthrough copy of SeqtoBlur.
  {
    const int n4 = (int)(szS / 4);
    copy_f4<<<n4 / 256, 256, 0, stream>>>((const float4*)SeqtoBlur, (float4*)outS, n4);
  }

  // Batched fp32 WMMA GEMM: avged_seq = S^T @ seq per batch.
  {
    dim3 grid(D_DIM / 64, T_DIM / 128, B_DIM);  // (8, 4, 256)
    blur_gemm_wmma<<<grid, 256, 0, stream>>>(SeqtoBlur, seq, outA);
  }

  // Ratio maps + avged_len.
  {
    const int n = B_DIM * T_DIM;
    ratios_kernel<<<n / 256, 256, 0, stream>>>(len_seq, avged_len, outR, outAR, outAL);
  }
}